// SemLevelGAT_67439576482331
// MI455X (gfx1250) — compile-verified
//
#include <hip/hip_runtime.h>
#include <hip/hip_bf16.h>

// SemLevelGAT on MI455X (gfx1250).
//
// Key simplification: softmax over a size-1 feature axis == 1.0, so the
// attention branch (edge_types, w_attn) never affects the output:
//   out = segment_sum(h[src], dst) @ W_lin^T
// Linear commutes with segment-sum, so we compute Wh = h @ W_lin^T once
// (fp32 WMMA 16x16x4 GEMM), then scatter-add Wh[src] into out[dst] with
// f32 global atomics. Wh (51.2 MB) and out (51.2 MB) both fit MI455X's
// 192 MB L2, so the random edge traffic is largely L2-resident.

typedef __attribute__((ext_vector_type(2))) float v2f;
typedef __attribute__((ext_vector_type(8))) float v8f;

#define DF 128   // D_FEAT == D_OUT == 128

// ---------------------------------------------------------------------------
// Zero the output buffer (segment_sum identity for nodes with no in-edges).
// ---------------------------------------------------------------------------
__global__ void zero_f4_kernel(float4* __restrict__ p, int n4) {
    int i = blockIdx.x * blockDim.x + threadIdx.x;
    if (i < n4) p[i] = make_float4(0.f, 0.f, 0.f, 0.f);
}

// ---------------------------------------------------------------------------
// Wh = h @ W_lin^T   (fp32, N x 128 @ 128 x 128)
// One wave per 16-row strip; 8 accumulator tiles cover all 128 columns.
// K loop in steps of 4 using V_WMMA_F32_16X16X4_F32 (8 independent WMMA
// chains in flight per K-step).
//
// VGPR layouts per CDNA5 ISA 7.12.2:
//   A (16x4 f32):  lane M = lane&15 ; {VGPR0,VGPR1} = K pair 2*(lane>>4)
//   B (4x16 f32):  lane N = lane&15 ; {VGPR0,VGPR1} = K pair 2*(lane>>4)
//   C (16x16 f32): VGPR v -> M = v + 8*(lane>>4), N = lane&15
// B[k][n] = W_lin[n*128 + k]  (W_lin stored [out,in] row-major).
// ---------------------------------------------------------------------------
__global__ __launch_bounds__(128) void wh_gemm_kernel(
        const float* __restrict__ h,
        const float* __restrict__ W,     // [128,128] row-major [out,in]
        float* __restrict__ Wh,
        int n_tiles)                      // number of 16-row tiles
{
    const int wave = threadIdx.x >> 5;                 // 0..3
    const int lane = threadIdx.x & 31;
    const int tile = blockIdx.x * 4 + wave;
    if (tile >= n_tiles) return;                       // whole wave exits: EXEC stays all-1s for WMMA
    const int row0 = tile * 16;
    const int half = lane >> 4;                        // selects K pair
    const int l15  = lane & 15;

    v8f c[8] = {};                                     // 8 tiles x 8 VGPRs

    const float* arow = h + (size_t)(row0 + l15) * DF + 2 * half;
    const float* bcol = W + (size_t)l15 * DF + 2 * half;   // + nt*16*DF per tile

    for (int k0 = 0; k0 < DF; k0 += 4) {
        v2f a;
        a.x = arow[k0 + 0];
        a.y = arow[k0 + 1];
#pragma unroll
        for (int nt = 0; nt < 8; ++nt) {
            const float* bp = bcol + (size_t)(nt * 16) * DF + k0;
            v2f b;
            b.x = bp[0];
            b.y = bp[1];
            c[nt] = __builtin_amdgcn_wmma_f32_16x16x4_f32(
                        /*neg_a=*/false, a, /*neg_b=*/false, b,
                        /*c_mod=*/(short)0, c[nt],
                        /*reuse_a=*/false, /*reuse_b=*/false);
        }
    }

#pragma unroll
    for (int nt = 0; nt < 8; ++nt) {
#pragma unroll
        for (int v = 0; v < 8; ++v) {
            const int r   = row0 + v + 8 * half;
            const int col = nt * 16 + l15;
            Wh[(size_t)r * DF + col] = c[nt][v];
        }
    }
}

// ---------------------------------------------------------------------------
// Edge scatter: one wave per edge; each lane moves 4 floats (128 = 32*4).
// out[dst] += Wh[src]  via f32 atomics (L2-resident destination).
// ---------------------------------------------------------------------------
__global__ __launch_bounds__(256) void edge_scatter_kernel(
        const float* __restrict__ Wh,
        const int* __restrict__ src,
        const int* __restrict__ dst,
        float* __restrict__ out,
        int n_edges)
{
    const int gw   = (blockIdx.x * blockDim.x + threadIdx.x) >> 5;  // edge id
    const int lane = threadIdx.x & 31;
    if (gw >= n_edges) return;
    const int s = src[gw];
    const int d = dst[gw];

    const float4 v = *(const float4*)(Wh + (size_t)s * DF + lane * 4);
    float* o = out + (size_t)d * DF + lane * 4;
    atomicAdd(o + 0, v.x);
    atomicAdd(o + 1, v.y);
    atomicAdd(o + 2, v.z);
    atomicAdd(o + 3, v.w);
}

// ---------------------------------------------------------------------------
extern "C" void kernel_launch(void* const* d_in, const int* in_sizes, int n_in,
                              void* d_out, int out_size, void* d_ws, size_t ws_size,
                              hipStream_t stream) {
    const float* h   = (const float*)d_in[0];   // [N,128]
    // d_in[1] = edge_types, d_in[2] = w_attn : mathematically dead (softmax
    // over a size-1 axis == 1.0), intentionally unused.
    const float* W   = (const float*)d_in[3];   // [128,128]
    const int*   src = (const int*)d_in[4];     // [E]
    const int*   dst = (const int*)d_in[5];     // [E]
    float*       out = (float*)d_out;           // [N,128]
    float*       Wh  = (float*)d_ws;            // scratch: N*128 floats (51.2 MB)

    const int n_nodes = in_sizes[0] / DF;
    const int n_edges = in_sizes[4];

    // 1) out = 0
    {
        const int n4 = out_size / 4;
        zero_f4_kernel<<<(n4 + 255) / 256, 256, 0, stream>>>((float4*)out, n4);
    }

    // 2) Wh = h @ W^T   (WMMA fp32)
    {
        const int n_tiles = (n_nodes + 15) / 16;        // 6250 (exact)
        const int blocks  = (n_tiles + 3) / 4;          // 4 waves/block
        wh_gemm_kernel<<<blocks, 128, 0, stream>>>(h, W, Wh, n_tiles);
    }

    // 3) out[dst] += Wh[src]
    {
        const long long waves  = (long long)n_edges;    // 1 wave per edge
        const int       blocks = (int)((waves * 32 + 255) / 256);
        edge_scatter_kernel<<<blocks, 256, 0, stream>>>(Wh, src, dst, out, n_edges);
    }
}